// NaiveVisCache_31920196944290
// MI455X (gfx1250) — compile-verified
//
#include <hip/hip_runtime.h>
#include <hip/hip_bf16.h>

// One thread processes 4 rays:
//   origins/dirs:  3x global_load_b128 each (perfectly coalesced, NT hint)
//   gathers:       4x global_load_b32 from the 48MB cache table (L2-resident)
//   output:        1x global_store_b128 (NT hint)
// Memory-bound: ~32 B/ray * 4M rays ~= 134 MB -> ~5.8 us at 23.3 TB/s.

typedef float v4f __attribute__((ext_vector_type(4)));

__device__ __forceinline__ unsigned expand_bits3(unsigned v) {
    v = (v | (v << 16)) & 0x030000FFu;
    v = (v | (v << 8))  & 0x0300F00Fu;
    v = (v | (v << 4))  & 0x030C30C3u;
    v = (v | (v << 2))  & 0x09249249u;
    return v;
}

__device__ __forceinline__ unsigned quant_coord(float o) {
    // (o/2 + 0.5) * 128, clip to [0,127], truncate (matches jnp astype(int32),
    // values are non-negative after clip so trunc == floor).
    // *0.5 and *128 are exact (powers of two) so fma contraction is harmless.
    float v = (o * 0.5f + 0.5f) * 128.0f;
    v = fminf(fmaxf(v, 0.0f), 127.0f);
    return (unsigned)(int)v;
}

__device__ __forceinline__ float one_ray(float ox, float oy, float oz,
                                         float dx, float dy, float dz,
                                         const int* __restrict__ cache) {
    float n = fmaxf(fmaxf(fabsf(dx), fabsf(dy)), fabsf(dz));
    // Reference: a = dx/n; conditions applied in order, LAST true one wins.
    // For n > 0 and round-to-nearest IEEE division, dx/n >= 1 <=> dx >= n and
    // dx/n <= -1 <=> dx <= -n, so we avoid the divides entirely.
    int face = 0;
    if (dx >=  n) face = 0;
    if (dx <= -n) face = 1;
    if (dy >=  n) face = 2;
    if (dy <= -n) face = 3;
    if (dz >=  n) face = 4;
    if (dz <= -n) face = 5;

    unsigned m = expand_bits3(quant_coord(ox))
               | (expand_bits3(quant_coord(oy)) << 1)
               | (expand_bits3(quant_coord(oz)) << 2);

    int val = __ldg(&cache[m * 6u + (unsigned)face]);
    return (val > 128) ? 1.0f : 0.0f;
}

__global__ __launch_bounds__(256) void vis_cache_kernel(
        const float* __restrict__ org,
        const float* __restrict__ dir,
        const int*   __restrict__ cache,
        float*       __restrict__ out,
        int nrays) {
    const unsigned t = blockIdx.x * blockDim.x + threadIdx.x;  // 4 rays / thread
    if ((int)(t * 4u) >= nrays) return;

    const v4f* o4 = (const v4f*)org;
    const v4f* d4 = (const v4f*)dir;

    // 12 floats = 4 rays worth of xyz, as three aligned 128-bit NT loads each.
    v4f oA = __builtin_nontemporal_load(&o4[3u * t + 0u]);
    v4f oB = __builtin_nontemporal_load(&o4[3u * t + 1u]);
    v4f oC = __builtin_nontemporal_load(&o4[3u * t + 2u]);
    v4f dA = __builtin_nontemporal_load(&d4[3u * t + 0u]);
    v4f dB = __builtin_nontemporal_load(&d4[3u * t + 1u]);
    v4f dC = __builtin_nontemporal_load(&d4[3u * t + 2u]);

    float of[12] = {oA[0], oA[1], oA[2], oA[3], oB[0], oB[1], oB[2], oB[3],
                    oC[0], oC[1], oC[2], oC[3]};
    float df[12] = {dA[0], dA[1], dA[2], dA[3], dB[0], dB[1], dB[2], dB[3],
                    dC[0], dC[1], dC[2], dC[3]};

    v4f res;
#pragma unroll
    for (int r = 0; r < 4; ++r) {
        res[r] = one_ray(of[3 * r + 0], of[3 * r + 1], of[3 * r + 2],
                         df[3 * r + 0], df[3 * r + 1], df[3 * r + 2], cache);
    }

    // Streaming output: NT store, keep L2 for the gather table.
    __builtin_nontemporal_store(res, &((v4f*)out)[t]);
}

extern "C" void kernel_launch(void* const* d_in, const int* in_sizes, int n_in,
                              void* d_out, int out_size, void* d_ws, size_t ws_size,
                              hipStream_t stream) {
    const float* org   = (const float*)d_in[0];  // (B,3) float32
    const float* dir   = (const float*)d_in[1];  // (B,3) float32
    const int*   cache = (const int*)d_in[2];    // (128^3, 6) int32
    float*       out   = (float*)d_out;          // B bools as float 0/1

    const int nrays    = out_size;               // 4194304
    const int nthreads = (nrays + 3) / 4;        // 4 rays per thread
    const int block    = 256;                    // 8 wave32 waves
    const int grid     = (nthreads + block - 1) / block;

    vis_cache_kernel<<<grid, block, 0, stream>>>(org, dir, cache, out, nrays);
}